// MultiBoxLoss_49658411877063
// MI455X (gfx1250) — compile-verified
//
#include <hip/hip_runtime.h>
#include <math.h>

#define KO 32          // number of GT boxes per batch
#define BLOCK 256
#define PPT 8          // priors per thread
#define CHUNK (BLOCK * PPT)   // 2048 priors per block

typedef __attribute__((ext_vector_type(2))) float v2f;
typedef __attribute__((ext_vector_type(8))) float v8f;

// Full-wave (32-lane) f32 sum using V_WMMA_F32_16X16X4_F32.
// A layout (16x4 f32): lanes 0-15 VGPR0 -> A[l][0]; lanes 16-31 VGPR0 -> A[l-16][2].
// So a.x = x for every lane places each lane's value exactly once.
// With B = all-ones, D[m][n] = x_m + x_{m+16}. Each lane sums its 8 D VGPRs
// (rows m..m+7 of its half), then one xor-shuffle across halves gives the total.
__device__ __forceinline__ float wave_sum_wmma(float x) {
  v2f a; a.x = x;    a.y = 0.0f;
  v2f b; b.x = 1.0f; b.y = 1.0f;
  v8f c = {};
  c = __builtin_amdgcn_wmma_f32_16x16x4_f32(false, a, false, b, (short)0, c, false, false);
  float s = c[0] + c[1] + c[2] + c[3] + c[4] + c[5] + c[6] + c[7];
  s += __shfl_xor(s, 16, 32);
  return s;
}

__device__ __forceinline__ float smooth_l1(float d) {
  float ad = fabsf(d);
  return (ad < 1.0f) ? 0.5f * d * d : ad - 0.5f;
}

// ---------------- init ----------------
__global__ void kInit(unsigned long long* packed, int* npos, int nPk, int B) {
  int i = blockIdx.x * BLOCK + threadIdx.x;
  if (i < nPk) packed[i] = 0ULL;
  if (i < B)  npos[i] = 0;
}

// ---------------- per-GT best prior (max IoU over P, first-index tie-break) ----
__global__ void kMatch(const float* __restrict__ priors,
                       const float* __restrict__ tboxes,
                       unsigned long long* __restrict__ packed, int P) {
  int b = blockIdx.y;
  __shared__ float4 gt[KO];
  __shared__ unsigned long long best[KO];
  int t = threadIdx.x;
  if (t < KO) {
    const float* g = tboxes + ((size_t)b * KO + t) * 4;
    gt[t] = make_float4(g[0], g[1], g[2], g[3]);
    best[t] = 0ULL;
  }
  __syncthreads();
  int base = blockIdx.x * CHUNK;
  for (int i = 0; i < PPT; ++i) {
    int p = base + i * BLOCK + t;
    if (p < P) {
      const float* pr = priors + (size_t)p * 4;
      float pcx = pr[0], pcy = pr[1], pw = pr[2], ph = pr[3];
      float pxmin = pcx - pw * 0.5f, pymin = pcy - ph * 0.5f;
      float pxmax = pcx + pw * 0.5f, pymax = pcy + ph * 0.5f;
      float parea = (pxmax - pxmin) * (pymax - pymin);
      unsigned int np = ~(unsigned int)p;   // larger = smaller p -> first-occurrence argmax
      for (int o = 0; o < KO; ++o) {
        float4 g = gt[o];
        float iw = fminf(g.z, pxmax) - fmaxf(g.x, pxmin);
        float ih = fminf(g.w, pymax) - fmaxf(g.y, pymin);
        iw = fmaxf(iw, 0.0f); ih = fmaxf(ih, 0.0f);
        float inter = iw * ih;
        float garea = (g.z - g.x) * (g.w - g.y);
        float iou = inter / (garea + parea - inter);
        unsigned long long pk = ((unsigned long long)__float_as_uint(iou) << 32) | np;
        atomicMax(&best[o], pk);
      }
    }
  }
  __syncthreads();
  if (t < KO) atomicMax(&packed[(size_t)b * KO + t], best[t]);
}

// ---------------- unpack per-GT results, validity mask ----------------
__global__ void kFixup(const unsigned long long* __restrict__ packed,
                       int* __restrict__ bp_idx, unsigned int* __restrict__ vmask) {
  int b = blockIdx.x;
  int o = threadIdx.x;   // 32 threads = 1 wave
  unsigned long long pk = packed[(size_t)b * KO + o];
  float ov = __uint_as_float((unsigned int)(pk >> 32));
  bp_idx[b * KO + o] = (int)(~(unsigned int)pk);
  bool valid = (ov >= 0.2f);
  unsigned long long bv = __ballot(valid);
  if (o == 0) vmask[b] = (unsigned int)bv;
}

// ---------------- main pass: match, smooth-L1, CE, mined ----------------
__global__ void kMain(const float* __restrict__ loc_data,
                      const float* __restrict__ conf_data,
                      const float* __restrict__ priors,
                      const float* __restrict__ tboxes,
                      const int* __restrict__ tlabels,
                      const int* __restrict__ bp_idx,
                      const unsigned int* __restrict__ vmask,
                      float* __restrict__ mined,
                      int* __restrict__ npos,
                      float* __restrict__ blk_lp,
                      float* __restrict__ blk_cep, int P) {
  int b = blockIdx.y;
  __shared__ float4 gt[KO];
  __shared__ int bpi[KO];
  __shared__ int lbl[KO];
  __shared__ float wsum[16];
  int t = threadIdx.x;
  if (t < KO) {
    const float* g = tboxes + ((size_t)b * KO + t) * 4;
    gt[t] = make_float4(g[0], g[1], g[2], g[3]);
    bpi[t] = bp_idx[b * KO + t];
    lbl[t] = tlabels[b * KO + t];
  }
  __syncthreads();
  unsigned int vm = vmask[b];
  bool any_valid = (vm != 0u);
  int base = blockIdx.x * CHUNK;
  float lp = 0.0f, cep = 0.0f;
  int np = 0;
  for (int i = 0; i < PPT; ++i) {
    int p = base + i * BLOCK + t;
    if (p < P) {
      const float* pr = priors + (size_t)p * 4;
      const float* cd = conf_data + ((size_t)b * P + p) * 2;
      __builtin_prefetch(cd + 2 * BLOCK, 0, 0);   // global_prefetch_b8
      float pcx = pr[0], pcy = pr[1], pw = pr[2], ph = pr[3];
      float pxmin = pcx - pw * 0.5f, pymin = pcy - ph * 0.5f;
      float pxmax = pcx + pw * 0.5f, pymax = pcy + ph * 0.5f;
      float parea = (pxmax - pxmin) * (pymax - pymin);
      // best truth over 32 GTs, first-occurrence argmax
      float bov = -1.0f; int bidx = 0;
      for (int o = 0; o < KO; ++o) {
        float4 g = gt[o];
        float iw = fminf(g.z, pxmax) - fmaxf(g.x, pxmin);
        float ih = fminf(g.w, pymax) - fmaxf(g.y, pymin);
        iw = fmaxf(iw, 0.0f); ih = fmaxf(ih, 0.0f);
        float inter = iw * ih;
        float garea = (g.z - g.x) * (g.w - g.y);
        float iou = inter / (garea + parea - inter);
        if (iou > bov) { bov = iou; bidx = o; }
      }
      // forced-match override (scatter with last-wins, matching .at[].set)
      int lasto = -1;
      for (int o = 0; o < KO; ++o) if (bpi[o] == p) lasto = o;
      if (lasto >= 0) {
        bidx = lasto;                              // idx set for ALL gts
        if ((vm >> lasto) & 1u) bov = 2.0f;        // valid -> 2.0; invalid -> original (no-op)
      }
      int conf = (bov < 0.35f) ? 0 : lbl[bidx];
      if (!any_valid) conf = 0;
      bool pos = (conf > 0);
      // cross entropy for 2 classes
      float c0 = cd[0], c1 = cd[1];
      float m = fmaxf(c0, c1);
      float lse = m + logf(expf(c0 - m) + expf(c1 - m));
      int cls = conf; cls = (cls < 0) ? 0 : ((cls > 1) ? 1 : cls);
      float ce = lse - (cls ? c1 : c0);
      mined[(size_t)b * P + p] = pos ? 0.0f : ce;
      if (pos) {
        cep += ce; np += 1;
        float4 g = gt[bidx];
        const float* ld = loc_data + ((size_t)b * P + p) * 4;
        float gcx = ((g.x + g.z) * 0.5f - pcx) / (0.1f * pw);
        float gcy = ((g.y + g.w) * 0.5f - pcy) / (0.1f * ph);
        float gw  = logf((g.z - g.x) / pw) / 0.2f;
        float gh  = logf((g.w - g.y) / ph) / 0.2f;
        lp += smooth_l1(ld[0] - gcx) + smooth_l1(ld[1] - gcy)
            + smooth_l1(ld[2] - gw)  + smooth_l1(ld[3] - gh);
      }
    }
  }
  // wave-level WMMA reductions (unconditional: EXEC all ones)
  float wlp  = wave_sum_wmma(lp);
  float wcep = wave_sum_wmma(cep);
  atomicAdd(&npos[b], np);                 // integer: order-independent, deterministic
  int wave = t >> 5;
  if ((t & 31) == 0) { wsum[wave] = wlp; wsum[8 + wave] = wcep; }
  __syncthreads();
  if (t == 0) {
    float a = 0.0f, c2 = 0.0f;
    for (int w = 0; w < 8; ++w) { a += wsum[w]; c2 += wsum[8 + w]; }
    blk_lp [b * gridDim.x + blockIdx.x] = a;
    blk_cep[b * gridDim.x + blockIdx.x] = c2;
  }
}

// ---------------- per-batch radix-select top-k sum + partial combine ----------
__global__ void kSelect(const float* __restrict__ mined,
                        const int* __restrict__ npos,
                        const float* __restrict__ blk_lp,
                        const float* __restrict__ blk_cep,
                        float* __restrict__ batch_lp,
                        float* __restrict__ batch_lc,
                        int P, int blocksPerBatch) {
  int b = blockIdx.x;
  int t = threadIdx.x;
  __shared__ unsigned int hist[256];
  __shared__ float red[256];
  __shared__ unsigned int s_prefix;
  __shared__ int s_krem;
  const unsigned int* vals = (const unsigned int*)(mined + (size_t)b * P);

  // in-order combine of block partials (deterministic)
  if (t == 0) {
    float lpsum = 0.0f, cepsum = 0.0f;
    for (int i = 0; i < blocksPerBatch; ++i) {
      lpsum  += blk_lp [b * blocksPerBatch + i];
      cepsum += blk_cep[b * blocksPerBatch + i];
    }
    batch_lp[b] = lpsum;
    red[0] = cepsum;
  }
  __syncthreads();
  float cep = red[0];
  __syncthreads();

  long long k64 = (long long)3 * (long long)npos[b];
  long long pm1 = (long long)P - 1;
  int k = (int)((k64 < pm1) ? k64 : pm1);

  float topk = 0.0f;
  if (k > 0) {
    unsigned int prefix = 0;
    int krem = k;
    for (int pass = 0; pass < 4; ++pass) {
      int shift = 24 - pass * 8;
      unsigned int himask = (pass == 0) ? 0u : (0xFFFFFFFFu << (shift + 8));
      hist[t] = 0;
      __syncthreads();
      for (int i = t; i < P; i += 256) {
        unsigned int v = vals[i];
        if ((v & himask) == prefix) atomicAdd(&hist[(v >> shift) & 255u], 1u);
      }
      __syncthreads();
      if (t == 0) {
        int cum = 0, bkt = 0;
        for (int bb = 255; bb >= 0; --bb) {
          int c = (int)hist[bb];
          if (cum + c >= krem) { bkt = bb; break; }
          cum += c;
        }
        s_prefix = prefix | ((unsigned int)bkt << shift);
        s_krem = krem - cum;
      }
      __syncthreads();
      prefix = s_prefix;
      krem = s_krem;
      __syncthreads();
    }
    unsigned int T = prefix;              // k-th largest value (bit pattern)
    float Tf = __uint_as_float(T);
    float part = 0.0f;
    for (int i = t; i < P; i += 256) {
      unsigned int v = vals[i];
      if (v > T) part += __uint_as_float(v);
    }
    red[t] = part;
    __syncthreads();
    for (int s = 128; s > 0; s >>= 1) {
      if (t < s) red[t] += red[t + s];
      __syncthreads();
    }
    if (t == 0) topk = red[0] + (float)krem * Tf;
  }
  if (t == 0) batch_lc[b] = cep + topk;
}

// ---------------- final scalar reduction ----------------
__global__ void kFinal(const float* __restrict__ batch_lp,
                       const float* __restrict__ batch_lc,
                       const int* __restrict__ npos,
                       float* __restrict__ out, int B) {
  if (threadIdx.x == 0 && blockIdx.x == 0) {
    float L = 0.0f, C = 0.0f;
    long long n = 0;
    for (int b = 0; b < B; ++b) { L += batch_lp[b]; C += batch_lc[b]; n += npos[b]; }
    float N = (float)n;
    if (N < 1.0f) N = 1.0f;
    out[0] = L / N;
    out[1] = C / N;
  }
}

extern "C" void kernel_launch(void* const* d_in, const int* in_sizes, int n_in,
                              void* d_out, int out_size, void* d_ws, size_t ws_size,
                              hipStream_t stream) {
  const float* loc    = (const float*)d_in[0];
  const float* conf   = (const float*)d_in[1];
  const float* priors = (const float*)d_in[2];
  const float* tbox   = (const float*)d_in[3];
  const int*   tlab   = (const int*)d_in[4];
  int P = in_sizes[2] / 4;     // priors: [P,4]
  int B = in_sizes[4] / KO;    // labels: [B,32]
  int gridx = (P + CHUNK - 1) / CHUNK;

  char* ws = (char*)d_ws;
  size_t off = 0;
  float* mined = (float*)(ws + off);                 off += (size_t)B * P * sizeof(float);
  unsigned long long* packed = (unsigned long long*)(ws + off); off += (size_t)B * KO * 8;
  int* bp_idx = (int*)(ws + off);                    off += (size_t)B * KO * 4;
  unsigned int* vmask = (unsigned int*)(ws + off);   off += (size_t)B * 4;
  int* npos = (int*)(ws + off);                      off += (size_t)B * 4;
  float* blk_lp = (float*)(ws + off);                off += (size_t)B * gridx * 4;
  float* blk_cep = (float*)(ws + off);               off += (size_t)B * gridx * 4;
  float* batch_lp = (float*)(ws + off);              off += (size_t)B * 4;
  float* batch_lc = (float*)(ws + off);              off += (size_t)B * 4;

  int nPk = B * KO;
  int ninit = (nPk > B) ? nPk : B;
  kInit<<<(ninit + BLOCK - 1) / BLOCK, BLOCK, 0, stream>>>(packed, npos, nPk, B);
  kMatch<<<dim3(gridx, B), BLOCK, 0, stream>>>(priors, tbox, packed, P);
  kFixup<<<B, KO, 0, stream>>>(packed, bp_idx, vmask);
  kMain<<<dim3(gridx, B), BLOCK, 0, stream>>>(loc, conf, priors, tbox, tlab,
                                              bp_idx, vmask, mined, npos,
                                              blk_lp, blk_cep, P);
  kSelect<<<B, 256, 0, stream>>>(mined, npos, blk_lp, blk_cep,
                                 batch_lp, batch_lc, P, gridx);
  kFinal<<<1, 32, 0, stream>>>(batch_lp, batch_lc, npos, (float*)d_out, B);
}